// VectorQuantization_58892591563252
// MI455X (gfx1250) — compile-verified
//
#include <hip/hip_runtime.h>

typedef __attribute__((ext_vector_type(2))) float v2f;
typedef __attribute__((ext_vector_type(8))) float v8f;

#define NTOK 16384   // b*n = 8*2048
#define NH   8
#define ND   64
#define NK   512

// ---------------------------------------------------------------------------
// Kernel 1: cluster assignment via fp32 WMMA (16x16x4).
// One wave per (32-token pair of 16x16 tiles, head): each loaded B tile (16
// clusters x 64 dims) feeds TWO WMMA chains, halving codebook re-traffic.
// Score = |m|^2 - 2*x.m  (|x|^2 is constant per token -> irrelevant to argmin).
// ---------------------------------------------------------------------------
__global__ __launch_bounds__(32) void vq_assign(const float* __restrict__ x,
                                                const float* __restrict__ means,
                                                float* __restrict__ ids_f,
                                                int*   __restrict__ ids_i)
{
    const int lane = threadIdx.x;      // one wave32 per block
    const int half = lane >> 4;        // 0 or 1
    const int l16  = lane & 15;
    // consecutive blocks share a head -> same 128KB codebook stream (WGP$ reuse)
    const int h    = blockIdx.x >> 9;            // NTOK/32 = 512 blocks per head
    const int tok0 = (blockIdx.x & 511) << 5;    // 32 tokens per wave

    // ---- A tiles: 2 x (16 tokens x 64 dims), pre-scaled by -2 -------------
    // fp32 16x4 A layout: lane(l16,half) holds [row=l16][4*kb + 2*half + j]
    const float* xrow0 = x + (size_t)(tok0 + l16)      * (NH * ND) + h * ND;
    const float* xrow1 = x + (size_t)(tok0 + 16 + l16) * (NH * ND) + h * ND;
    v2f a0[16], a1[16];
#pragma unroll
    for (int kb = 0; kb < 16; ++kb) {
        v2f t0 = *(const v2f*)(xrow0 + kb * 4 + 2 * half);
        v2f t1 = *(const v2f*)(xrow1 + kb * 4 + 2 * half);
        a0[kb].x = -2.0f * t0.x;  a0[kb].y = -2.0f * t0.y;
        a1[kb].x = -2.0f * t1.x;  a1[kb].y = -2.0f * t1.y;
    }

    float minv[2][8];
    int   mini[2][8];
#pragma unroll
    for (int t = 0; t < 2; ++t)
#pragma unroll
        for (int r = 0; r < 8; ++r) { minv[t][r] = 3.4e38f; mini[t][r] = 0; }

    const float* mh = means + (size_t)h * NK * ND;

    for (int tile = 0; tile < NK / 16; ++tile) {
        const float* mrow = mh + (size_t)(tile * 16 + l16) * ND;
        if (tile + 1 < NK / 16)
            __builtin_prefetch(mh + (size_t)((tile + 1) * 16 + l16) * ND, 0, 1);

        // ---- B tile: 16 clusters x 64 dims; also accumulate |m|^2 --------
        v2f bm[16];
        float ss = 0.0f;
#pragma unroll
        for (int kb = 0; kb < 16; ++kb) {
            bm[kb] = *(const v2f*)(mrow + kb * 4 + 2 * half);
            ss += bm[kb].x * bm[kb].x + bm[kb].y * bm[kb].y;
        }
        // each lane covered 32 dims of cluster l16; partner half has the rest
        const float msq = ss + __shfl_xor(ss, 16, 32);

        // C starts as |m|^2 broadcast per cluster column; A already holds -2x
        v8f c0, c1;
#pragma unroll
        for (int r = 0; r < 8; ++r) { c0[r] = msq; c1[r] = msq; }

#pragma unroll
        for (int kb = 0; kb < 16; ++kb) {
            c0 = __builtin_amdgcn_wmma_f32_16x16x4_f32(
                     false, a0[kb], false, bm[kb], (short)0, c0, false, false);
            c1 = __builtin_amdgcn_wmma_f32_16x16x4_f32(
                     false, a1[kb], false, bm[kb], (short)0, c1, false, false);
        }

        const int cid = tile * 16 + l16;
#pragma unroll
        for (int r = 0; r < 8; ++r) {
            if (c0[r] < minv[0][r]) { minv[0][r] = c0[r]; mini[0][r] = cid; }
            if (c1[r] < minv[1][r]) { minv[1][r] = c1[r]; mini[1][r] = cid; }
        }
    }

    // ---- argmin across the 16 lanes of each half (ties -> lower index) ----
#pragma unroll
    for (int off = 1; off < 16; off <<= 1) {
#pragma unroll
        for (int t = 0; t < 2; ++t)
#pragma unroll
            for (int r = 0; r < 8; ++r) {
                float ov = __shfl_xor(minv[t][r], off, 32);
                int   oi = __shfl_xor(mini[t][r], off, 32);
                if (ov < minv[t][r] || (ov == minv[t][r] && oi < mini[t][r])) {
                    minv[t][r] = ov; mini[t][r] = oi;
                }
            }
    }

    // D layout: vgpr r holds token M = r + 8*half
    if (l16 == 0) {
#pragma unroll
        for (int t = 0; t < 2; ++t)
#pragma unroll
            for (int r = 0; r < 8; ++r) {
                const int tok = tok0 + t * 16 + half * 8 + r;
                ids_f[(size_t)tok * NH + h] = (float)mini[t][r];
                ids_i[(size_t)tok * NH + h] = mini[t][r];
            }
    }
}

// ---------------------------------------------------------------------------
// Kernel 2: scatter EMA statistics. One wave per (token, head); lane d covers
// dims d and d+32 with global fp32 atomics into the ~1MB L2-resident scratch.
// ---------------------------------------------------------------------------
__global__ void vq_scatter(const float* __restrict__ x,
                           const int*   __restrict__ ids,
                           float* __restrict__ cnt,
                           float* __restrict__ sums)
{
    const int gid  = blockIdx.x * blockDim.x + threadIdx.x;
    const int wid  = gid >> 5;
    const int lane = gid & 31;
    if (wid >= NTOK * NH) return;
    const int tok = wid >> 3;
    const int h   = wid & 7;
    const int id  = ids[wid];

    const float* xp = x + (size_t)tok * (NH * ND) + h * ND;
    float* sp = sums + (size_t)(h * NK + id) * ND;
    atomicAdd(sp + lane,      xp[lane]);
    atomicAdd(sp + lane + 32, xp[lane + 32]);
    if (lane == 0) atomicAdd(cnt + h * NK + id, 1.0f);
}

// ---------------------------------------------------------------------------
// Kernel 3: updated_means = 0.001 * sum/(eps+cnt) + 0.999 * means
// ---------------------------------------------------------------------------
__global__ void vq_update(const float* __restrict__ means,
                          const float* __restrict__ cnt,
                          const float* __restrict__ sums,
                          float* __restrict__ out_means)
{
    const int i = blockIdx.x * blockDim.x + threadIdx.x;
    if (i >= NH * NK * ND) return;
    const int hk = i >> 6;
    const float nm = sums[i] / (1e-6f + cnt[hk]);
    out_means[i] = 0.999f * means[i] + 0.001f * nm;
}

extern "C" void kernel_launch(void* const* d_in, const int* in_sizes, int n_in,
                              void* d_out, int out_size, void* d_ws, size_t ws_size,
                              hipStream_t stream)
{
    const float* x     = (const float*)d_in[0];   // [8,2048,512] fp32
    const float* means = (const float*)d_in[1];   // [8,512,64]   fp32
    float* out = (float*)d_out;                   // [131072 ids | 262144 means]

    float* cnt  = (float*)d_ws;                   // [NH*NK]
    float* sums = cnt + NH * NK;                  // [NH*NK*ND]
    int*   idsi = (int*)(sums + (size_t)NH * NK * ND); // [NTOK*NH]

    hipMemsetAsync(d_ws, 0, (size_t)(NH * NK + NH * NK * ND) * sizeof(float), stream);

    vq_assign <<< dim3((NTOK / 32) * NH), dim3(32), 0, stream >>> (x, means, out, idsi);
    vq_scatter<<< dim3((NTOK * NH) / 8), dim3(256), 0, stream >>> (x, idsi, cnt, sums);
    vq_update <<< dim3((NH * NK * ND + 255) / 256), dim3(256), 0, stream >>> (
        means, cnt, sums, out + (size_t)NTOK * NH);
}